// GraphSAGEEncoder_65781719105819
// MI455X (gfx1250) — compile-verified
//
#include <hip/hip_runtime.h>

#define DIM 128

typedef __attribute__((ext_vector_type(2))) float v2f;
typedef __attribute__((ext_vector_type(8))) float v8f;

// ---------------------------------------------------------------------------
// Zero a float buffer with float4 stores (counts here are multiples of 4).
// ---------------------------------------------------------------------------
__global__ void zero_kernel(float* __restrict__ p, long n) {
  long i = ((long)blockIdx.x * blockDim.x + threadIdx.x) * 4;
  if (i + 3 < n) {
    *(float4*)(p + i) = make_float4(0.f, 0.f, 0.f, 0.f);
  } else {
    for (; i < n; ++i) p[i] = 0.0f;
  }
}

// ---------------------------------------------------------------------------
// Repack a 128x128 row-major weight W[k][n] into K-pair-interleaved form:
//   Wp[((k>>1)*DIM + n)*2 + 0] = W[k][n], [..+1] = W[k+1][n]   (k even)
// so a WMMA B fragment (needs W[k][col], W[k+1][col]) is ONE b64 load.
// ---------------------------------------------------------------------------
__global__ void repack_w_kernel(const float* __restrict__ W,
                                float* __restrict__ Wp) {
  int i = blockIdx.x * blockDim.x + threadIdx.x;  // 0 .. DIM/2*DIM-1
  if (i >= (DIM / 2) * DIM) return;
  int kk = i / DIM;  // k/2
  int n = i % DIM;
  float2 v;
  v.x = W[(size_t)(2 * kk) * DIM + n];
  v.y = W[(size_t)(2 * kk + 1) * DIM + n];
  *(float2*)(Wp + (size_t)i * 2) = v;
}

// ---------------------------------------------------------------------------
// Edge scatter: sums[dst] += h[src] (float4 per thread, 32 threads / edge),
// optionally cnt[dst] += 1. Atomics stay L2-resident (51MB dest << 192MB L2).
// Index loads are same-address within a wave -> merged broadcast requests.
// ---------------------------------------------------------------------------
__global__ void scatter_kernel(const float* __restrict__ h,
                               const long long* __restrict__ ei,
                               float* __restrict__ sums,
                               float* __restrict__ cnt,
                               int nedges, int add_cnt) {
  long gid = (long)blockIdx.x * blockDim.x + threadIdx.x;
  int e = (int)(gid >> 5);
  if (e >= nedges) return;
  int c = (int)(gid & 31);
  long long s = ei[e];                 // edge_index[0, e]
  long long d = ei[(long)nedges + e];  // edge_index[1, e]
  const float4 v = *(const float4*)(h + (size_t)s * DIM + (size_t)c * 4);
  float* dstp = sums + (size_t)d * DIM + (size_t)c * 4;
  atomicAdd(dstp + 0, v.x);
  atomicAdd(dstp + 1, v.y);
  atomicAdd(dstp + 2, v.z);
  atomicAdd(dstp + 3, v.w);
  if (add_cnt && c == 0) atomicAdd(cnt + d, 1.0f);
}

// ---------------------------------------------------------------------------
// Fused SAGE layer: out = relu((sums/max(cnt,1)) @ Wl + b + xin @ Wr)
// via v_wmma_f32_16x16x4_f32 (fp32 WMMA, full precision).
//
// Wave tile = 32 rows x 16 cols (two stacked 16x16 tiles, 4 accumulators).
// B fragments (bL,bR) are loaded ONCE per k-step and reused by both row
// tiles: 6 b64 loads : 4 WMMAs per k-step (vs 8:4 with a single tile).
// Block = 256 threads = 8 waves -> 32 rows x 128 cols.
//
// Per-row 1/cnt mean scaling is algebraically deferred to the epilogue
// (scaling A row M == scaling D row M), keeping the loop pure load+WMMA and
// all consecutive WMMAs independent (4 distinct accumulators).
//
// A 16x4 f32 fragment (v2f): lanes 0-15 -> M=lane,    K=k0+{0,1};
//                            lanes 16-31 -> M=lane-16, K=k0+{2,3}.
// B 4x16 f32 fragment (v2f): lanes 0-15 -> N=lane,    K=k0+{0,1};
//                            lanes 16-31 -> N=lane-16, K=k0+{2,3}.
// C/D 16x16 f32 (v8f): VGPR v -> row base+v (lanes 0-15) / base+8+v (16-31),
//                      col n0 + (lane&15).
// ---------------------------------------------------------------------------
__global__ __launch_bounds__(256) void sage_wmma_kernel(
    const float* __restrict__ sums, const float* __restrict__ cnt,
    const float* __restrict__ xin,
    const float* __restrict__ WlP, const float* __restrict__ WrP,
    const float* __restrict__ bias, float* __restrict__ out, int nrows) {
  const int lane = threadIdx.x & 31;
  const int wave = threadIdx.x >> 5;
  const int m0 = blockIdx.x << 5;       // 32-row super-tile base
  const int n0 = wave << 4;             // 16-col tile base (8 waves -> 128)
  const int l15 = lane & 15;
  int row0 = m0 + l15;                  // tile 0 rows
  int row1 = m0 + 16 + l15;             // tile 1 rows
  row0 = row0 < nrows ? row0 : nrows - 1;  // branch-free clamps: EXEC full
  row1 = row1 < nrows ? row1 : nrows - 1;
  const int koff = (lane >> 4) << 1;    // 0 for lanes 0-15, 2 for lanes 16-31
  const int col = n0 + l15;

  const float bv = bias[col];
  v8f accL0 = { 0.f, 0.f, 0.f, 0.f, 0.f, 0.f, 0.f, 0.f };
  v8f accL1 = { 0.f, 0.f, 0.f, 0.f, 0.f, 0.f, 0.f, 0.f };
  v8f accR0 = { bv, bv, bv, bv, bv, bv, bv, bv };
  v8f accR1 = { bv, bv, bv, bv, bv, bv, bv, bv };

  const float* al0 = sums + (size_t)row0 * DIM;
  const float* ar0 = xin + (size_t)row0 * DIM;
  const float* al1 = sums + (size_t)row1 * DIM;
  const float* ar1 = xin + (size_t)row1 * DIM;

#pragma unroll
  for (int k0 = 0; k0 < DIM; k0 += 4) {
    const int k = k0 + koff;  // even
    // B fragments: shared across both row tiles
    v2f bL = *(const v2f*)(WlP + ((size_t)(k >> 1) * DIM + col) * 2);
    v2f bR = *(const v2f*)(WrP + ((size_t)(k >> 1) * DIM + col) * 2);
    // A fragments: 2 matrices x 2 row tiles
    v2f aL0 = *(const v2f*)(al0 + k);
    v2f aR0 = *(const v2f*)(ar0 + k);
    v2f aL1 = *(const v2f*)(al1 + k);
    v2f aR1 = *(const v2f*)(ar1 + k);
    accL0 = __builtin_amdgcn_wmma_f32_16x16x4_f32(
        false, aL0, false, bL, (short)0, accL0, false, false);
    accR0 = __builtin_amdgcn_wmma_f32_16x16x4_f32(
        false, aR0, false, bR, (short)0, accR0, false, false);
    accL1 = __builtin_amdgcn_wmma_f32_16x16x4_f32(
        false, aL1, false, bL, (short)0, accL1, false, false);
    accR1 = __builtin_amdgcn_wmma_f32_16x16x4_f32(
        false, aR1, false, bR, (short)0, accR1, false, false);
  }

  // Epilogue: per-output-row mean scale, add paths, ReLU, store (both tiles).
  const int mb = (lane >> 4) << 3;  // 0 or 8
#pragma unroll
  for (int v = 0; v < 8; ++v) {
    const int mrow0 = m0 + mb + v;
    if (mrow0 < nrows) {
      const float rc = 1.0f / fmaxf(cnt[mrow0], 1.0f);  // broadcast load
      float r = accL0[v] * rc + accR0[v];
      out[(size_t)mrow0 * DIM + col] = r > 0.0f ? r : 0.0f;
    }
    const int mrow1 = m0 + 16 + mb + v;
    if (mrow1 < nrows) {
      const float rc = 1.0f / fmaxf(cnt[mrow1], 1.0f);
      float r = accL1[v] * rc + accR1[v];
      out[(size_t)mrow1 * DIM + col] = r > 0.0f ? r : 0.0f;
    }
  }
}

// ---------------------------------------------------------------------------
// Inputs (setup_inputs order):
//   0: x [N,128] f32      1: edge_index [2,E] i64
//   2: W_l0 [128,128]     3: b_l0 [128]     4: W_r0 [128,128]
//   5: W_l1 [128,128]     6: b_l1 [128]     7: W_r1 [128,128]
// Output: h2 [N,128] f32.
// Workspace: [ sums N*128 | cnt N | h N*128 | Wl0p 16K | Wr0p 16K |
//              Wl1p 16K | Wr1p 16K ]  (~103.3 MB)
// ---------------------------------------------------------------------------
extern "C" void kernel_launch(void* const* d_in, const int* in_sizes, int n_in,
                              void* d_out, int out_size, void* d_ws, size_t ws_size,
                              hipStream_t stream) {
  (void)n_in; (void)out_size; (void)ws_size;
  const float* x      = (const float*)d_in[0];
  const long long* ei = (const long long*)d_in[1];
  const float* W_l0   = (const float*)d_in[2];
  const float* b_l0   = (const float*)d_in[3];
  const float* W_r0   = (const float*)d_in[4];
  const float* W_l1   = (const float*)d_in[5];
  const float* b_l1   = (const float*)d_in[6];
  const float* W_r1   = (const float*)d_in[7];

  const int nnodes = in_sizes[0] / DIM;  // 100000
  const int nedges = in_sizes[1] / 2;    // 600000

  float* sums = (float*)d_ws;
  float* cnt  = sums + (size_t)nnodes * DIM;
  float* h    = cnt + nnodes;
  float* wl0p = h + (size_t)nnodes * DIM;
  float* wr0p = wl0p + DIM * DIM;
  float* wl1p = wr0p + DIM * DIM;
  float* wr1p = wl1p + DIM * DIM;
  float* outp = (float*)d_out;

  const long scatThreads = (long)nedges * 32;
  const int scatBlocks = (int)((scatThreads + 255) / 256);
  const int gemmBlocks = (nnodes + 31) / 32;               // 32-row supertiles
  const int repackBlocks = ((DIM / 2) * DIM + 255) / 256;  // 32

  // One-shot weight repack (tiny)
  repack_w_kernel<<<repackBlocks, 256, 0, stream>>>(W_l0, wl0p);
  repack_w_kernel<<<repackBlocks, 256, 0, stream>>>(W_r0, wr0p);
  repack_w_kernel<<<repackBlocks, 256, 0, stream>>>(W_l1, wl1p);
  repack_w_kernel<<<repackBlocks, 256, 0, stream>>>(W_r1, wr1p);

  // ---- Layer 0 ----
  const long z0 = (long)nnodes * DIM + nnodes;  // sums + cnt contiguous
  zero_kernel<<<(int)((z0 / 4 + 255) / 256), 256, 0, stream>>>(sums, z0);
  scatter_kernel<<<scatBlocks, 256, 0, stream>>>(x, ei, sums, cnt, nedges, 1);
  sage_wmma_kernel<<<gemmBlocks, 256, 0, stream>>>(sums, cnt, x, wl0p, wr0p,
                                                   b_l0, h, nnodes);

  // ---- Layer 1 (degrees cnt are identical; only re-zero sums) ----
  const long z1 = (long)nnodes * DIM;
  zero_kernel<<<(int)((z1 / 4 + 255) / 256), 256, 0, stream>>>(sums, z1);
  scatter_kernel<<<scatBlocks, 256, 0, stream>>>(h, ei, sums, cnt, nedges, 0);
  sage_wmma_kernel<<<gemmBlocks, 256, 0, stream>>>(sums, cnt, h, wl1p, wr1p,
                                                   b_l1, outp, nnodes);
}